// EventMessagePassingNode_38740605010510
// MI455X (gfx1250) — compile-verified
//
#include <hip/hip_runtime.h>

typedef float v2f __attribute__((ext_vector_type(2)));
typedef float v8f __attribute__((ext_vector_type(8)));

#define N_NODES 50000
#define N_EDGES 600000
#define DFEAT   128

// ---------------------------------------------------------------------------
// Kernel 1: zero the aggregation workspace (N_NODES x 128 f32 in d_ws)
// ---------------------------------------------------------------------------
__global__ void zero_agg_kernel(float4* __restrict__ ws, int n4) {
    int i = blockIdx.x * blockDim.x + threadIdx.x;
    if (i < n4) ws[i] = make_float4(0.f, 0.f, 0.f, 0.f);
}

// ---------------------------------------------------------------------------
// Kernel 2: per-edge message + scatter-add.
// One wave32 per edge; lane owns 4 contiguous features (128 = 32*4).
// msg = h[dst[e]] * e_h[e];  agg[dst[e]] += msg  (f32 global atomics, L2-resident)
// ---------------------------------------------------------------------------
__global__ void scatter_mul_kernel(const float* __restrict__ h,
                                   const float* __restrict__ e_h,
                                   const int*   __restrict__ dst,
                                   float*       __restrict__ agg) {
    int gid  = blockIdx.x * blockDim.x + threadIdx.x;
    int edge = gid >> 5;
    int lane = gid & 31;
    if (edge >= N_EDGES) return;

    int d = dst[edge];
    int c = lane * 4;

    const float4 hv = *(const float4*)(h   + (size_t)d    * DFEAT + c);
    const float4 ev = *(const float4*)(e_h + (size_t)edge * DFEAT + c);

    float* outp = agg + (size_t)d * DFEAT + c;
    atomicAdd(outp + 0, hv.x * ev.x);
    atomicAdd(outp + 1, hv.y * ev.y);
    atomicAdd(outp + 2, hv.z * ev.z);
    atomicAdd(outp + 3, hv.w * ev.w);
}

// ---------------------------------------------------------------------------
// Kernel 3: out = (agg @ W^T + b) * norm  via V_WMMA_F32_16X16X4_F32.
// Block = 256 threads (8 waves). blockIdx.x = 16-node tile (50000 = 3125*16).
// Wave w computes the 16x16 output tile for feature columns [16w, 16w+16).
//
// f32 16x4 A layout (ISA 7.12.2): lane (r, half), VGPR j holds
//   A[M=r][K = 2*half + j]; B (4x16) is the mirrored layout:
//   lane (r, half), VGPR j holds B[K = 2*half + j][N=r].
// Since B[k][n] = W[n][k], lane (r,half) preloads W[(n0+r)*128 + 4*kk + 2*half + {0,1}].
// C/D layout: VGPR v, lane (r, half) -> row (v + 8*half), col r.
// ---------------------------------------------------------------------------
#define LDS_STRIDE 132   // 128 + 4 pad: r*132 % 64 = 4r -> conflict-free A reads

__launch_bounds__(256)
__global__ void gemm_norm_kernel(const float* __restrict__ agg,
                                 const float* __restrict__ W,
                                 const float* __restrict__ bias,
                                 const float* __restrict__ norm,
                                 float*       __restrict__ out) {
    __shared__ float sA[16 * LDS_STRIDE];

    const int m0  = blockIdx.x * 16;
    const int tid = threadIdx.x;

    // Cooperative stage of the 16x128 agg tile: 256 threads x 8 floats.
    {
        int row = tid >> 4;           // 0..15
        int col = (tid & 15) * 8;     // 0,8,...,120
        const float* src = agg + (size_t)(m0 + row) * DFEAT + col;
        float*       dss = sA + row * LDS_STRIDE + col;
#pragma unroll
        for (int i = 0; i < 8; ++i) dss[i] = src[i];
    }
    __syncthreads();

    const int wave = tid >> 5;        // 0..7 -> feature tile
    const int lane = tid & 31;
    const int half = lane >> 4;       // 0 or 1
    const int r    = lane & 15;       // 0..15
    const int n0   = wave * 16;

    // Preload this wave's 16x128 block of W into registers (64 VGPRs/lane).
    const float* wbase = W + (size_t)(n0 + r) * DFEAT + 2 * half;
    v2f breg[32];
#pragma unroll
    for (int kk = 0; kk < 32; ++kk)
        breg[kk] = *(const v2f*)(wbase + 4 * kk);

    const float* abase = sA + r * LDS_STRIDE + 2 * half;  // 8B-aligned
    v8f c = {};
#pragma unroll
    for (int kk = 0; kk < 32; ++kk) {
        v2f a = *(const v2f*)(abase + 4 * kk);
        // 8 args: (neg_a, A, neg_b, B, c_mod, C, reuse_a, reuse_b)
        c = __builtin_amdgcn_wmma_f32_16x16x4_f32(false, a, false, breg[kk],
                                                  (short)0, c, false, false);
    }

    // Epilogue: bias + norm scaling, direct store from the C/D layout.
    const float bcol = bias[n0 + r];
#pragma unroll
    for (int v = 0; v < 8; ++v) {
        int row = m0 + v + 8 * half;
        out[(size_t)row * DFEAT + n0 + r] = (c[v] + bcol) * norm[row];
    }
}

// ---------------------------------------------------------------------------
extern "C" void kernel_launch(void* const* d_in, const int* in_sizes, int n_in,
                              void* d_out, int out_size, void* d_ws, size_t ws_size,
                              hipStream_t stream) {
    const float* h    = (const float*)d_in[0];   // [N_NODES, 128]
    const float* e_h  = (const float*)d_in[1];   // [N_EDGES, 128]
    const float* norm = (const float*)d_in[2];   // [N_NODES, 1]
    const int*   dst  = (const int*)  d_in[3];   // [N_EDGES]
    const float* W    = (const float*)d_in[4];   // [128, 128]
    const float* b    = (const float*)d_in[5];   // [128]
    float* out = (float*)d_out;                  // [N_NODES, 128]
    float* agg = (float*)d_ws;                   // [N_NODES, 128] scratch

    // 1) agg = 0
    const int n4 = N_NODES * DFEAT / 4;
    zero_agg_kernel<<<(n4 + 255) / 256, 256, 0, stream>>>((float4*)agg, n4);

    // 2) scatter-add messages (1 wave / edge)
    const long long threads = (long long)N_EDGES * 32;
    scatter_mul_kernel<<<(int)((threads + 255) / 256), 256, 0, stream>>>(h, e_h, dst, agg);

    // 3) WMMA GEMM + bias + norm (50000 = 3125 * 16, exact)
    gemm_norm_kernel<<<N_NODES / 16, 256, 0, stream>>>(agg, W, b, norm, out);
}